// Model_18769007083597
// MI455X (gfx1250) — compile-verified
//
#include <hip/hip_runtime.h>
#include <cstddef>

// ---------------------------------------------------------------------------
// SpiralNet mesh autoencoder on MI455X (gfx1250), fp32 throughout.
// GEMMs on V_WMMA_F32_16X16X4_F32 (wave32).  Weights are pre-transposed to
// K-major (K x COP, COP = COUT padded to 16, zero-filled) so every B-fragment
// load is a coalesced 64B b32 load across the 16 lanes of a column tile.
// Pooling (both directions) is an exact 3-tap gather (rows = repeat(arange)).
// ---------------------------------------------------------------------------

typedef float v2f __attribute__((ext_vector_type(2)));
typedef float v8f __attribute__((ext_vector_type(8)));

__device__ __forceinline__ v8f wmma_f32x4(v2f a, v2f b, v8f c) {
    return __builtin_amdgcn_wmma_f32_16x16x4_f32(false, a, false, b, (short)0, c, false, false);
}

__device__ __forceinline__ float elu_f(float x)     { return x > 0.0f ? x : expm1f(x); }
__device__ __forceinline__ float sigmoid_f(float x) { return 1.0f / (1.0f + expf(-x)); }

// ---------------------------------------------------------------------------
// Weight transpose + pad:  W (CO x K, row-major)  ->  Wt (K x COP), cols >= CO zero.
// ---------------------------------------------------------------------------
__global__ __launch_bounds__(256) void transpose_w(const float* __restrict__ W,
                                                   float* __restrict__ Wt,
                                                   int CO, int K, int COP)
{
    int tid = blockIdx.x * 256 + threadIdx.x;
    if (tid >= K * COP) return;
    int k = tid / COP, o = tid % COP;
    Wt[tid] = (o < CO) ? W[(size_t)o * K + k] : 0.0f;
}

// Build transposed+padded level-0 weight: Wpt (36 x 32) from enW0 (32 x 27).
// K-slot kk = 4*s + c ; c < 3 real, c == 3 zero.
__global__ __launch_bounds__(256) void build_wpad_t(const float* __restrict__ W,
                                                    float* __restrict__ Wpt)
{
    int tid = blockIdx.x * 256 + threadIdx.x;
    if (tid >= 36 * 32) return;
    int kk = tid / 32, o = tid % 32;
    int s = kk >> 2, c = kk & 3;
    Wpt[tid] = (c < 3) ? W[o * 27 + s * 3 + c] : 0.0f;
}

// ---------------------------------------------------------------------------
// Generic spiral conv: out[b,n,o] = act( sum_k g[b,n,k] * W[o,k] + bias[o] )
// g[b,n, s*CIN+c] = h[b, si[n,s], c].  CIN in {32,64}, COUT in {3,32,64}.
// Wt is K-major (K x COP).  One wave = one 16-row tile x all column tiles.
// ---------------------------------------------------------------------------
template <int CIN, int COUT, bool DO_ELU>
__global__ __launch_bounds__(256) void spiral_conv_wmma(
    const float* __restrict__ h, const int* __restrict__ si,
    const float* __restrict__ Wt, const float* __restrict__ bias,
    float* __restrict__ out, int Nn, int logNn)
{
    constexpr int COP = (COUT + 15) & ~15;
    constexpr int NT  = COP / 16;

    const int lane = threadIdx.x & 31;
    const int wid  = threadIdx.x >> 5;
    const int tile = blockIdx.x * 8 + wid;
    const int r0   = tile << 4;
    const int b    = r0 >> logNn;          // 16-row tiles never straddle a batch
    const int nb   = r0 & (Nn - 1);
    const int M    = lane & 15;            // A-row / B-col / D-col index
    const int kh   = lane >> 4;            // K half-select
    const int n    = nb + M;

    const float* hb  = h + (((size_t)b << logNn) * CIN);
    const float* wto = Wt + M;             // + t*16 per column tile

    v8f acc[NT];
#pragma unroll
    for (int t = 0; t < NT; ++t) acc[t] = (v8f)(0.0f);

    const int* sip = si + n * 9;
    for (int s = 0; s < 9; ++s) {
        const int ns = sip[s];
        const float* hp = hb + (size_t)ns * CIN;
        if (s < 8) __builtin_prefetch(hb + (size_t)sip[s + 1] * CIN, 0, 0);
        const int kgs = s * CIN;
#pragma unroll
        for (int cb = 0; cb < CIN; cb += 4) {
            const int koff = cb + 2 * kh;                  // even -> 8B aligned
            v2f a = *(const v2f*)(hp + koff);
            const float* wp = wto + (size_t)(kgs + koff) * COP;
#pragma unroll
            for (int t = 0; t < NT; ++t) {
                v2f bf;
                bf.x = wp[t * 16];                         // B[kg  ][o] coalesced
                bf.y = wp[t * 16 + COP];                   // B[kg+1][o] coalesced
                acc[t] = wmma_f32x4(a, bf, acc[t]);
            }
        }
    }

    const size_t obase = ((size_t)b << logNn) * COUT;
#pragma unroll
    for (int t = 0; t < NT; ++t) {
        const int o = t * 16 + M;
        const bool ov = (COUT % 16 == 0) || (o < COUT);
        const float bv = ov ? bias[o] : 0.0f;
#pragma unroll
        for (int v = 0; v < 8; ++v) {
            if (ov) {
                const int row = nb + v + 8 * kh;           // D: VGPR v holds M = v + 8*half
                float x = acc[t][v] + bv;
                if (DO_ELU) x = elu_f(x);
                out[obase + (size_t)row * COUT + o] = x;
            }
        }
    }
}

// ---------------------------------------------------------------------------
// Level-0 conv: CIN=3, K=27 padded to 36 (4 slots per tap, 4th = 0),
// consuming transposed padded weight Wpt (36 x 32).
// ---------------------------------------------------------------------------
__global__ __launch_bounds__(256) void spiral_conv_c3_wmma(
    const float* __restrict__ x, const int* __restrict__ si,
    const float* __restrict__ Wpt, const float* __restrict__ bias,
    float* __restrict__ out, int Nn, int logNn)
{
    constexpr int COUT = 32, NT = 2;
    const int lane = threadIdx.x & 31;
    const int wid  = threadIdx.x >> 5;
    const int tile = blockIdx.x * 8 + wid;
    const int r0   = tile << 4;
    const int b    = r0 >> logNn;
    const int nb   = r0 & (Nn - 1);
    const int M    = lane & 15;
    const int kh   = lane >> 4;
    const int n    = nb + M;

    const float* hb  = x + (((size_t)b << logNn) * 3);
    const float* wto = Wpt + M;
    const float ay   = (kh == 0) ? 1.0f : 0.0f;   // half1's 2nd slot is the pad zero

    v8f acc[NT];
#pragma unroll
    for (int t = 0; t < NT; ++t) acc[t] = (v8f)(0.0f);

    const int* sip = si + n * 9;
    for (int s = 0; s < 9; ++s) {
        const int ns = sip[s];
        const float* hp = hb + (size_t)ns * 3;
        v2f a;
        a.x = hp[2 * kh];                          // c = 0 (half0) or c = 2 (half1)
        a.y = hp[1] * ay;                          // unconditional load, masked (no branch)
        const float* wp = wto + (size_t)(4 * s + 2 * kh) * 32;
#pragma unroll
        for (int t = 0; t < NT; ++t) {
            v2f bf;
            bf.x = wp[t * 16];
            bf.y = wp[t * 16 + 32];
            acc[t] = wmma_f32x4(a, bf, acc[t]);
        }
    }

    const size_t obase = ((size_t)b << logNn) * COUT;
#pragma unroll
    for (int t = 0; t < NT; ++t) {
        const int o = t * 16 + M;
        const float bv = bias[o];
#pragma unroll
        for (int v = 0; v < 8; ++v) {
            const int row = nb + v + 8 * kh;
            out[obase + (size_t)row * COUT + o] = elu_f(acc[t][v] + bv);
        }
    }
}

// ---------------------------------------------------------------------------
// Dense GEMM, R == 16 rows exactly (batch fills the M tile), Wt K-major (K x CO):
// out[r,o] = act( X[r,:] . W[o,:] + bias[o] ).  One wave per 16-column tile.
// out2/out3: optional duplicate destinations (fc writes z and mu slots).
// ---------------------------------------------------------------------------
template <int ACT> // 0 = none, 1 = sigmoid
__global__ __launch_bounds__(256) void dense_gemm_wmma(
    const float* __restrict__ X, const float* __restrict__ Wt,
    const float* __restrict__ bias, float* __restrict__ out,
    float* __restrict__ out2, float* __restrict__ out3, int K, int CO)
{
    const int lane = threadIdx.x & 31;
    const int wid  = threadIdx.x >> 5;
    const int ct   = blockIdx.x * 8 + wid;
    const int M    = lane & 15;
    const int kh   = lane >> 4;
    const int o    = ct * 16 + M;

    const float* xr = X + (size_t)M * K;
    const float* wc = Wt + o;

    v8f acc = (v8f)(0.0f);
    for (int k = 0; k < K; k += 4) {
        const int kg = k + 2 * kh;
        v2f a = *(const v2f*)(xr + kg);
        v2f bf;
        bf.x = wc[(size_t)kg * CO];
        bf.y = wc[(size_t)(kg + 1) * CO];
        acc = wmma_f32x4(a, bf, acc);
    }

    const float bv = bias[o];
#pragma unroll
    for (int v = 0; v < 8; ++v) {
        const int r = v + 8 * kh;
        float val = acc[v] + bv;
        if (ACT == 1) val = sigmoid_f(val);
        const size_t oi = (size_t)r * CO + o;
        out[oi] = val;
        if (out2) out2[oi] = val;
        if (out3) out3[oi] = val;
    }
}

// ---------------------------------------------------------------------------
// Pooling (both directions): rows = repeat(arange(Nout),3) =>
// out[b,n,c] = sum_{j<3} val[3n+j] * in[b, col[3n+j], c].  Pure gather.
// ---------------------------------------------------------------------------
template <int C>
__global__ __launch_bounds__(256) void pool_kernel(
    const float* __restrict__ in, const int* __restrict__ col,
    const float* __restrict__ val, float* __restrict__ out,
    int Nout, int logNout, int Nin)
{
    constexpr int CLOG = (C == 32) ? 5 : 6;
    const int idx  = blockIdx.x * 256 + threadIdx.x;
    const int c    = idx & (C - 1);
    const int rest = idx >> CLOG;
    const int n    = rest & (Nout - 1);
    const int b    = rest >> logNout;

    const float* inb = in + (size_t)b * Nin * C;
    float s = 0.0f;
#pragma unroll
    for (int j = 0; j < 3; ++j) {
        const int   cc = col[3 * n + j];
        const float vv = val[3 * n + j];
        s += vv * inb[(size_t)cc * C + c];
    }
    out[idx] = s;
}

// ---------------------------------------------------------------------------
// Host-side orchestration
// ---------------------------------------------------------------------------
extern "C" void kernel_launch(void* const* d_in, const int* in_sizes, int n_in,
                              void* d_out, int out_size, void* d_ws, size_t ws_size,
                              hipStream_t stream)
{
    (void)in_sizes; (void)n_in; (void)out_size; (void)ws_size;

    // Input map (setup_inputs dict order)
    const float* x    = (const float*)d_in[0];
    const int*   si0  = (const int*)d_in[1];
    const int*   si1  = (const int*)d_in[2];
    const int*   si2  = (const int*)d_in[3];
    const int*   si3  = (const int*)d_in[4];
    const int*   dcol[4] = { (const int*)d_in[6],  (const int*)d_in[12],
                             (const int*)d_in[18], (const int*)d_in[24] };
    const float* dval[4] = { (const float*)d_in[7],  (const float*)d_in[13],
                             (const float*)d_in[19], (const float*)d_in[25] };
    const int*   ucol[4] = { (const int*)d_in[9],  (const int*)d_in[15],
                             (const int*)d_in[21], (const int*)d_in[27] };
    const float* uval[4] = { (const float*)d_in[10], (const float*)d_in[16],
                             (const float*)d_in[22], (const float*)d_in[28] };
    const float* enW0 = (const float*)d_in[29];
    const float* enb0 = (const float*)d_in[30];
    const float* enW1 = (const float*)d_in[31];
    const float* enb1 = (const float*)d_in[32];
    const float* enW2 = (const float*)d_in[33];
    const float* enb2 = (const float*)d_in[34];
    const float* enW3 = (const float*)d_in[35];
    const float* enb3 = (const float*)d_in[36];
    const float* fcW  = (const float*)d_in[37];
    const float* fcb  = (const float*)d_in[38];
    const float* dfcW = (const float*)d_in[39];
    const float* dfcb = (const float*)d_in[40];
    const float* deW0 = (const float*)d_in[41];
    const float* deb0 = (const float*)d_in[42];
    const float* deW1 = (const float*)d_in[43];
    const float* deb1 = (const float*)d_in[44];
    const float* deW2 = (const float*)d_in[45];
    const float* deb2 = (const float*)d_in[46];
    const float* deW3 = (const float*)d_in[47];
    const float* deb3 = (const float*)d_in[48];
    const float* outW = (const float*)d_in[49];
    const float* outb = (const float*)d_in[50];

    float* outF  = (float*)d_out;
    float* zOut  = outF + (size_t)16 * 32768 * 3;        // 1,572,864
    float* muOut = zOut + 16 * 256;

    // Workspace: 2 x 64MB ping-pong + 8MB transposed-weight slab + small (~142.7 MB)
    const size_t F0 = (size_t)16 * 32768 * 32;
    float* A    = (float*)d_ws;
    float* Bf   = A + F0;
    float* Wt   = Bf + F0;                 // up to 2,097,152 floats (reused per layer)
    float* Wpt  = Wt + 2097152;            // 1152 floats
    float* Zlat = Wpt + 1280;              // 4096 floats

    dim3 blk(256);
    auto convBlocks = [](int Nn) { return (16 * Nn / 16) / 8; };
    auto tBlocks    = [](int total) { return (total + 255) / 256; };

    // --- encoder ---
    build_wpad_t<<<dim3(tBlocks(36 * 32)), blk, 0, stream>>>(enW0, Wpt);
    spiral_conv_c3_wmma<<<dim3(convBlocks(32768)), blk, 0, stream>>>(x, si0, Wpt, enb0, A, 32768, 15);
    pool_kernel<32><<<dim3(16 * 8192 * 32 / 256), blk, 0, stream>>>(A, dcol[0], dval[0], Bf, 8192, 13, 32768);

    transpose_w<<<dim3(tBlocks(288 * 32)), blk, 0, stream>>>(enW1, Wt, 32, 288, 32);
    spiral_conv_wmma<32, 32, true><<<dim3(convBlocks(8192)), blk, 0, stream>>>(Bf, si1, Wt, enb1, A, 8192, 13);
    pool_kernel<32><<<dim3(16 * 2048 * 32 / 256), blk, 0, stream>>>(A, dcol[1], dval[1], Bf, 2048, 11, 8192);

    transpose_w<<<dim3(tBlocks(288 * 32)), blk, 0, stream>>>(enW2, Wt, 32, 288, 32);
    spiral_conv_wmma<32, 32, true><<<dim3(convBlocks(2048)), blk, 0, stream>>>(Bf, si2, Wt, enb2, A, 2048, 11);
    pool_kernel<32><<<dim3(16 * 512 * 32 / 256), blk, 0, stream>>>(A, dcol[2], dval[2], Bf, 512, 9, 2048);

    transpose_w<<<dim3(tBlocks(288 * 64)), blk, 0, stream>>>(enW3, Wt, 64, 288, 64);
    spiral_conv_wmma<32, 64, true><<<dim3(convBlocks(512)), blk, 0, stream>>>(Bf, si3, Wt, enb3, A, 512, 9);
    pool_kernel<64><<<dim3(16 * 128 * 64 / 256), blk, 0, stream>>>(A, dcol[3], dval[3], Bf, 128, 7, 512);

    // --- bottleneck: fc (sigmoid -> z == mu), dfc ---
    transpose_w<<<dim3(tBlocks(8192 * 256)), blk, 0, stream>>>(fcW, Wt, 256, 8192, 256);
    dense_gemm_wmma<1><<<dim3(2), blk, 0, stream>>>(Bf, Wt, fcb, Zlat, zOut, muOut, 8192, 256);
    transpose_w<<<dim3(tBlocks(256 * 8192)), blk, 0, stream>>>(dfcW, Wt, 8192, 256, 8192);
    dense_gemm_wmma<0><<<dim3(64), blk, 0, stream>>>(Zlat, Wt, dfcb, A, nullptr, nullptr, 256, 8192);

    // --- decoder: up-pool then conv+ELU ---
    pool_kernel<64><<<dim3(16 * 512 * 64 / 256), blk, 0, stream>>>(A, ucol[3], uval[3], Bf, 512, 9, 128);
    transpose_w<<<dim3(tBlocks(576 * 64)), blk, 0, stream>>>(deW0, Wt, 64, 576, 64);
    spiral_conv_wmma<64, 64, true><<<dim3(convBlocks(512)), blk, 0, stream>>>(Bf, si3, Wt, deb0, A, 512, 9);

    pool_kernel<64><<<dim3(16 * 2048 * 64 / 256), blk, 0, stream>>>(A, ucol[2], uval[2], Bf, 2048, 11, 512);
    transpose_w<<<dim3(tBlocks(576 * 32)), blk, 0, stream>>>(deW1, Wt, 32, 576, 32);
    spiral_conv_wmma<64, 32, true><<<dim3(convBlocks(2048)), blk, 0, stream>>>(Bf, si2, Wt, deb1, A, 2048, 11);

    pool_kernel<32><<<dim3(16 * 8192 * 32 / 256), blk, 0, stream>>>(A, ucol[1], uval[1], Bf, 8192, 13, 2048);
    transpose_w<<<dim3(tBlocks(288 * 32)), blk, 0, stream>>>(deW2, Wt, 32, 288, 32);
    spiral_conv_wmma<32, 32, true><<<dim3(convBlocks(8192)), blk, 0, stream>>>(Bf, si1, Wt, deb2, A, 8192, 13);

    pool_kernel<32><<<dim3(16 * 32768 * 32 / 256), blk, 0, stream>>>(A, ucol[0], uval[0], Bf, 32768, 15, 8192);
    transpose_w<<<dim3(tBlocks(288 * 32)), blk, 0, stream>>>(deW3, Wt, 32, 288, 32);
    spiral_conv_wmma<32, 32, true><<<dim3(convBlocks(32768)), blk, 0, stream>>>(Bf, si0, Wt, deb3, A, 32768, 15);

    // --- output head (no activation), straight into d_out ---
    transpose_w<<<dim3(tBlocks(288 * 16)), blk, 0, stream>>>(outW, Wt, 3, 288, 16);
    spiral_conv_wmma<32, 3, false><<<dim3(convBlocks(32768)), blk, 0, stream>>>(A, si0, Wt, outb, outF, 32768, 15);
}